// KVCache_498216206780
// MI455X (gfx1250) — compile-verified
//
#include <hip/hip_runtime.h>

// KV-cache sink+window eviction: pure streaming copy, bandwidth-bound (~1.07 GB
// HBM traffic, ~46us at 23.3 TB/s). Split into a divisionless bulk kernel for
// the shifted window and a tiny edge kernel for sink + new rows.
//
// out[t<4]         = C[t]
// out[4<=t<4084]   = C[t+12]
// out[4084<=t<4100]= C_new[t-4084]
// C=[4,32,4096,128] f32, C_new=[4,32,16,128] f32, out=[4,32,4100,128] f32 (x2: K,V)

typedef __attribute__((ext_vector_type(4))) float v4f;

namespace {
constexpr int T         = 4096;
constexpr int T_NEW     = 16;
constexpr int D         = 128;
constexpr int WINDOW    = 4096;
constexpr int SINK      = 4;
constexpr int T_OUT     = SINK + WINDOW;              // 4100
constexpr int BH        = 4 * 32;                     // 128 (b,h) slices per tensor
constexpr int NEW_START = T_OUT - T_NEW;              // 4084
constexpr int SHIFT     = T + T_NEW - WINDOW - SINK;  // 12
constexpr long ROWS_PER_TENSOR = (long)BH * T_OUT;    // 524,800
constexpr int ROWS_PER_WAVE  = 4;                     // bulk: 4 rows/wave
constexpr int WAVES_PER_BLK  = 4;                     // bulk: 128-thread blocks
constexpr int ROWS_PER_BLK   = ROWS_PER_WAVE * WAVES_PER_BLK;   // 16
constexpr int WINDOW_ROWS    = NEW_START - SINK;      // 4080 = 255 * 16 exactly
constexpr int BULK_GRID_X    = WINDOW_ROWS / ROWS_PER_BLK;      // 255
constexpr int EDGE_ROWS      = SINK + T_NEW;          // 20 rows per slice
}

// ---------------------------------------------------------------------------
// Bulk: out[:, :, 4:4084, :] = C[:, :, 16:4096, :]
// grid=(255, 256), block=128. blockIdx.y = tensor*128 + bh. No div, no branch.
// ---------------------------------------------------------------------------
__global__ __launch_bounds__(WAVES_PER_BLK * 32)
void kv_window_copy(const float* __restrict__ K,
                    const float* __restrict__ V,
                    float* __restrict__ out)
{
    const int wave   = threadIdx.x >> 5;
    const int lane   = threadIdx.x & 31;
    const int slice  = blockIdx.y;            // 0..255
    const int bh     = slice & (BH - 1);
    const int tensor = slice >> 7;            // 0 = K, 1 = V

    const int t0 = SINK + blockIdx.x * ROWS_PER_BLK + wave * ROWS_PER_WAVE;

    const float* __restrict__ src =
        (tensor ? V : K) + ((long)bh * T + (t0 + SHIFT)) * D + lane * 4;
    float* __restrict__ dst =
        out + (long)tensor * (ROWS_PER_TENSOR * D)
            + ((long)bh * T_OUT + t0) * D + lane * 4;

    // 4 outstanding b128 NT loads per lane, then 4 b128 NT stores.
    v4f vals[ROWS_PER_WAVE];
    #pragma unroll
    for (int j = 0; j < ROWS_PER_WAVE; ++j)
        vals[j] = __builtin_nontemporal_load((const v4f*)(src + j * D));

    #pragma unroll
    for (int j = 0; j < ROWS_PER_WAVE; ++j)
        __builtin_nontemporal_store(vals[j], (v4f*)(dst + j * D));
}

// ---------------------------------------------------------------------------
// Edges: sink rows (t<4 from C) + new rows (t>=4084 from C_new).
// grid=(4, 256), block=160 (5 waves). One row per wave; branch is wave-uniform.
// ---------------------------------------------------------------------------
__global__ __launch_bounds__(160)
void kv_edge_copy(const float* __restrict__ K,
                  const float* __restrict__ V,
                  const float* __restrict__ Kn,
                  const float* __restrict__ Vn,
                  float* __restrict__ out)
{
    const int wave   = threadIdx.x >> 5;      // 0..4
    const int lane   = threadIdx.x & 31;
    const int slice  = blockIdx.y;            // 0..255
    const int bh     = slice & (BH - 1);
    const int tensor = slice >> 7;

    const int tl = blockIdx.x * 5 + wave;     // 0..19 local edge row

    const float* __restrict__ src;
    int t;
    if (tl < SINK) {                          // sink: copy C row tl
        t   = tl;
        src = (tensor ? V : K) + ((long)bh * T + tl) * D;
    } else {                                  // new: copy C_new row tl-4
        const int tn = tl - SINK;
        t   = NEW_START + tn;
        src = (tensor ? Vn : Kn) + ((long)bh * T_NEW + tn) * D;
    }

    float* __restrict__ dst =
        out + (long)tensor * (ROWS_PER_TENSOR * D)
            + ((long)bh * T_OUT + t) * D;

    v4f val = __builtin_nontemporal_load((const v4f*)src + lane);
    __builtin_nontemporal_store(val, (v4f*)dst + lane);
}

extern "C" void kernel_launch(void* const* d_in, const int* in_sizes, int n_in,
                              void* d_out, int out_size, void* d_ws, size_t ws_size,
                              hipStream_t stream)
{
    (void)in_sizes; (void)n_in; (void)out_size; (void)d_ws; (void)ws_size;

    const float* K  = (const float*)d_in[0];
    const float* V  = (const float*)d_in[1];
    const float* Kn = (const float*)d_in[2];
    const float* Vn = (const float*)d_in[3];
    float* out = (float*)d_out;

    // Bulk window copy: 255*256 blocks x 128 threads, 16 rows/block (exact).
    kv_window_copy<<<dim3(BULK_GRID_X, 2 * BH), dim3(WAVES_PER_BLK * 32), 0, stream>>>(
        K, V, out);

    // Edge copy: 4*256 blocks x 160 threads, 5 rows/block (exact, 20 rows/slice).
    kv_edge_copy<<<dim3(EDGE_ROWS / 5, 2 * BH), dim3(160), 0, stream>>>(
        K, V, Kn, Vn, out);
}